// LogMarginalLikelihood_14207751815447
// MI455X (gfx1250) — compile-verified
//
#include <hip/hip_runtime.h>
#include <math.h>

typedef __attribute__((ext_vector_type(2))) float v2f;
typedef __attribute__((ext_vector_type(8))) float v8f;
typedef __attribute__((ext_vector_type(4))) int   v4i;

#define AS_GLOBAL __attribute__((address_space(1)))
#define AS_LDS    __attribute__((address_space(3)))

#define NN     8192
#define MREAL  101
#define MPAD   112      // 7 tiles of 16 columns; cols 101..111 are zero padding
#define TPROBE 100
#define PITER  30
#define KC     32       // k-chunk staged in LDS (double-buffered)
#define ROWS   64       // rows per workgroup (4 waves x 16 rows)
#define TPB    128      // 4 waves
#define SPLITK 4
#define KSPLIT (NN / SPLITK)
#define NCH    (KSPLIT / KC)
#define LDT    36       // padded LDS stride (dwords): 16B-aligned, 36%8==4 -> conflict-free b64

#if defined(__HIP_DEVICE_COMPILE__) && \
    __has_builtin(__builtin_amdgcn_global_load_async_to_lds_b128) && \
    __has_builtin(__builtin_amdgcn_s_wait_asynccnt)
#define USE_ASYNC_LDS 1
#else
#define USE_ASYNC_LDS 0
#endif

// ---------------- block reduction helper (wave32-agnostic) ----------------
__device__ __forceinline__ float block_reduce_256(float v) {
    __shared__ float sm[256];
    __syncthreads();
    sm[threadIdx.x] = v;
    __syncthreads();
    for (int s = 128; s > 0; s >>= 1) {
        if ((int)threadIdx.x < s) sm[threadIdx.x] += sm[threadIdx.x + s];
        __syncthreads();
    }
    return sm[0];
}

// ---------------- init: B = [y, Z], X=0, R=P=B, rs = colsum(B*B) ----------
__global__ __launch_bounds__(256) void gp_init(
    const float* __restrict__ y, const float* __restrict__ Z,
    float* __restrict__ X, float* __restrict__ R, float* __restrict__ P,
    float* __restrict__ rs)
{
    const int m = blockIdx.x;            // 0..MPAD-1
    float acc = 0.f;
    for (int i = threadIdx.x; i < NN; i += 256) {
        float b = 0.f;
        if (m == 0)           b = y[i];
        else if (m <= TPROBE) b = Z[(size_t)i * TPROBE + (m - 1)];
        X[(size_t)m * NN + i] = 0.f;
        R[(size_t)m * NN + i] = b;
        P[(size_t)m * NN + i] = b;
        acc += b * b;
    }
    float tot = block_reduce_256(acc);
    if (threadIdx.x == 0) rs[m] = tot;
}

// ---------------- V = K @ P via fp32 WMMA, split-K partials ----------------
// Per workgroup: 64 rows x all 112 columns over a 2048-wide k-split.
// K streamed from HBM exactly once per CG iteration; P tiles are L2-resident.
// Double-buffered memory->LDS async staging overlaps HBM with WMMA compute.
__global__ __launch_bounds__(TPB, 1) void gp_gemm_wmma(
    const float* __restrict__ Kmat, const float* __restrict__ P,
    float* __restrict__ Vp)
{
    __shared__ __align__(16) float ldsK[2][ROWS * LDT];   // [buf][row][k]
    __shared__ __align__(16) float ldsP[2][MPAD * LDT];   // [buf][col][k]

    const int tid  = threadIdx.x;
    const int wave = tid >> 5;
    const int lane = tid & 31;
    const int rb   = blockIdx.x * ROWS;     // global row base
    const int ks   = blockIdx.y * KSPLIT;   // k range base for this split

    v8f acc[7];
    #pragma unroll
    for (int c = 0; c < 7; ++c) {
        #pragma unroll
        for (int j = 0; j < 8; ++j) acc[c][j] = 0.0f;
    }

    const int lrow = lane & 15;             // A-row / B-col within tile
    const int kb   = (lane >> 4) << 1;      // lanes 0-15: k{0,1}; 16-31: k{2,3}
    const int arow = wave * 16 + lrow;      // this wave's LDS row

#if USE_ASYNC_LDS
    // ---- async double-buffered pipeline -----------------------------------
    auto stage = [&](int kc, int buf) {
        // K tile: 64 rows x 32 k = 512 float4 -> 4 async b128 per thread
        #pragma unroll
        for (int j = 0; j < 4; ++j) {
            int idx = tid + j * TPB;
            int r   = idx >> 3;
            int c4  = (idx & 7) << 2;
            const float* g = Kmat + (size_t)(rb + r) * NN + (ks + kc + c4);
            __builtin_amdgcn_global_load_async_to_lds_b128(
                (AS_GLOBAL v4i*)g,
                (AS_LDS v4i*)&ldsK[buf][r * LDT + c4],
                0, 0);
        }
        // P tile: 112 cols x 32 k = 896 float4 -> 7 async b128 per thread
        #pragma unroll
        for (int j = 0; j < 7; ++j) {
            int idx = tid + j * TPB;
            int c   = idx >> 3;
            int k4  = (idx & 7) << 2;
            const float* g = P + (size_t)c * NN + (ks + kc + k4);
            __builtin_amdgcn_global_load_async_to_lds_b128(
                (AS_GLOBAL v4i*)g,
                (AS_LDS v4i*)&ldsP[buf][c * LDT + k4],
                0, 0);
        }
    };

    stage(0, 0);
    for (int c = 0; c < NCH; ++c) {
        if (c + 1 < NCH) {
            stage((c + 1) * KC, (c + 1) & 1);       // prefetch next chunk
            __builtin_amdgcn_s_wait_asynccnt(11);   // allow next chunk in flight
        } else {
            __builtin_amdgcn_s_wait_asynccnt(0);
        }
        __syncthreads();
        const int buf = c & 1;
        #pragma unroll
        for (int kk = 0; kk < KC; kk += 4) {
            v2f a = *(const v2f*)(&ldsK[buf][arow * LDT + kk + kb]);
            #pragma unroll
            for (int ct = 0; ct < 7; ++ct) {
                v2f b = *(const v2f*)(&ldsP[buf][(ct * 16 + lrow) * LDT + kk + kb]);
                acc[ct] = __builtin_amdgcn_wmma_f32_16x16x4_f32(
                    false, a, false, b, (short)0, acc[ct], false, false);
            }
        }
        __syncthreads();   // all reads done before this buf is refilled
    }
#else
    // ---- fallback: synchronous staging ------------------------------------
    for (int c = 0; c < NCH; ++c) {
        const int kc  = c * KC;
        const int buf = c & 1;
        __syncthreads();
        #pragma unroll
        for (int j = 0; j < 4; ++j) {
            int idx = tid + j * TPB;
            int r   = idx >> 3;
            int c4  = (idx & 7) << 2;
            float4 kv = *(const float4*)(Kmat + (size_t)(rb + r) * NN + (ks + kc + c4));
            *(float4*)(&ldsK[buf][r * LDT + c4]) = kv;
        }
        #pragma unroll
        for (int j = 0; j < 7; ++j) {
            int idx = tid + j * TPB;
            int cc  = idx >> 3;
            int k4  = (idx & 7) << 2;
            float4 pv4 = *(const float4*)(P + (size_t)cc * NN + (ks + kc + k4));
            *(float4*)(&ldsP[buf][cc * LDT + k4]) = pv4;
        }
        __syncthreads();
        #pragma unroll
        for (int kk = 0; kk < KC; kk += 4) {
            v2f a = *(const v2f*)(&ldsK[buf][arow * LDT + kk + kb]);
            #pragma unroll
            for (int ct = 0; ct < 7; ++ct) {
                v2f b = *(const v2f*)(&ldsP[buf][(ct * 16 + lrow) * LDT + kk + kb]);
                acc[ct] = __builtin_amdgcn_wmma_f32_16x16x4_f32(
                    false, a, false, b, (short)0, acc[ct], false, false);
            }
        }
    }
#endif

    // store partial: Vp[split][col][row]; C/D layout: vgpr j -> M=j(+8), N=lane&15
    float* out = Vp + (size_t)blockIdx.y * ((size_t)MPAD * NN);
    const int m0  = (lane >> 4) * 8;
    const int col = lane & 15;
    #pragma unroll
    for (int ct = 0; ct < 7; ++ct) {
        #pragma unroll
        for (int j = 0; j < 8; ++j) {
            out[(size_t)(ct * 16 + col) * NN + (rb + wave * 16 + m0 + j)] = acc[ct][j];
        }
    }
}

// -------- reduce split-K partials into V and compute pv = colsum(P*V) -----
__global__ __launch_bounds__(256) void gp_pv(
    const float* __restrict__ Vp, const float* __restrict__ P,
    float* __restrict__ V, float* __restrict__ pv)
{
    const int m = blockIdx.x;
    const size_t stride = (size_t)MPAD * NN;
    float acc = 0.f;
    for (int i = threadIdx.x; i < NN; i += 256) {
        size_t off = (size_t)m * NN + i;
        float v = Vp[off] + Vp[stride + off] + Vp[2 * stride + off] + Vp[3 * stride + off];
        V[off] = v;
        acc += P[off] * v;
    }
    float tot = block_reduce_256(acc);
    if (threadIdx.x == 0) pv[m] = tot;
}

// -------- X += a*P; R -= a*V; rs_new = colsum(R*R); record alpha ----------
__global__ __launch_bounds__(256) void gp_xr(
    float* __restrict__ X, float* __restrict__ R,
    const float* __restrict__ P, const float* __restrict__ V,
    const float* __restrict__ rs, const float* __restrict__ pv,
    float* __restrict__ rs_new, float* __restrict__ alphas, int iter)
{
    const int m = blockIdx.x;
    const float d = pv[m];
    const float a = (d != 0.f) ? rs[m] / d : 0.f;   // padded cols stay zero
    float acc = 0.f;
    for (int i = threadIdx.x; i < NN; i += 256) {
        size_t off = (size_t)m * NN + i;
        X[off] += a * P[off];
        float r = R[off] - a * V[off];
        R[off] = r;
        acc += r * r;
    }
    float tot = block_reduce_256(acc);
    if (threadIdx.x == 0) {
        rs_new[m] = tot;
        alphas[iter * MPAD + m] = a;
    }
}

// -------- P = R + beta*P; rs = rs_new; record beta ------------------------
__global__ __launch_bounds__(256) void gp_pupd(
    float* __restrict__ P, const float* __restrict__ R,
    float* __restrict__ rs, const float* __restrict__ rs_new,
    float* __restrict__ betas, int iter)
{
    const int m = blockIdx.x;
    const float ro = rs[m];
    const float b  = (ro != 0.f) ? rs_new[m] / ro : 0.f;
    for (int i = threadIdx.x; i < NN; i += 256) {
        size_t off = (size_t)m * NN + i;
        P[off] = R[off] + b * P[off];
    }
    __syncthreads();   // all reads of rs[m] done before thread 0 overwrites
    if (threadIdx.x == 0) {
        betas[iter * MPAD + m] = b;
        rs[m] = rs_new[m];
    }
}

// -------- SLQ: per probe, tridiagonal eigensolve tracking first row -------
__global__ __launch_bounds__(128) void gp_slq(
    const float* __restrict__ alphas, const float* __restrict__ betas,
    float* __restrict__ quad)
{
    const int t = blockIdx.x * blockDim.x + threadIdx.x;
    if (t >= TPROBE) return;
    const int m = t + 1;              // skip y column
    const int n = PITER;

    float d[PITER], e[PITER], z[PITER];
    for (int i = 0; i < n; ++i) {
        float ai = alphas[i * MPAD + m];
        float di = 1.f / ai;
        if (i > 0) di += betas[(i - 1) * MPAD + m] / alphas[(i - 1) * MPAD + m];
        d[i] = di;
        z[i] = (i == 0) ? 1.f : 0.f;
    }
    for (int i = 0; i < n - 1; ++i)
        e[i] = sqrtf(betas[i * MPAD + m]) / alphas[i * MPAD + m];
    e[n - 1] = 0.f;

    // implicit-shift QL (tqli), accumulating only first eigenvector row in z
    for (int l = 0; l < n; ++l) {
        int iter = 0;
        int mm;
        do {
            for (mm = l; mm < n - 1; ++mm) {
                float dd = fabsf(d[mm]) + fabsf(d[mm + 1]);
                if (fabsf(e[mm]) <= 1.1920929e-07f * dd) break;
            }
            if (mm != l) {
                if (++iter > 60) break;
                float g = (d[l + 1] - d[l]) / (2.f * e[l]);
                float r = hypotf(g, 1.f);
                g = d[mm] - d[l] + e[l] / (g + copysignf(r, g));
                float s = 1.f, c = 1.f, p = 0.f;
                bool underflow = false;
                for (int i = mm - 1; i >= l; --i) {
                    float f = s * e[i];
                    float b = c * e[i];
                    r = hypotf(f, g);
                    e[i + 1] = r;
                    if (r == 0.f) { d[i + 1] -= p; e[mm] = 0.f; underflow = true; break; }
                    s = f / r; c = g / r;
                    g = d[i + 1] - p;
                    r = (d[i] - g) * s + 2.f * c * b;
                    p = s * r;
                    d[i + 1] = g + p;
                    g = c * r - b;
                    f = z[i + 1];
                    z[i + 1] = s * z[i] + c * f;
                    z[i]     = c * z[i] - s * f;
                }
                if (!underflow) { d[l] -= p; e[l] = g; e[mm] = 0.f; }
            }
        } while (mm != l);
    }

    float q = 0.f;
    for (int i = 0; i < n; ++i) {
        float lam = fmaxf(d[i], 1e-12f);
        q += z[i] * z[i] * logf(lam);
    }
    quad[t] = q;
}

// -------- final combine: -0.5*y'X0 - 0.5*N*mean(quad) - N/2*log(2pi) ------
__global__ __launch_bounds__(256) void gp_final(
    const float* __restrict__ y, const float* __restrict__ X,
    const float* __restrict__ quad, float* __restrict__ out)
{
    float acc = 0.f;
    for (int i = threadIdx.x; i < NN; i += 256) acc += y[i] * X[i];  // X col 0
    float dot = block_reduce_256(acc);
    float q = (threadIdx.x < TPROBE) ? quad[threadIdx.x] : 0.f;
    float qsum = block_reduce_256(q);
    if (threadIdx.x == 0) {
        float logdet = (float)NN * (qsum / (float)TPROBE);
        out[0] = -0.5f * dot - 0.5f * logdet
                 - (float)NN * 0.5f * logf(6.283185307179586f);
    }
}

extern "C" void kernel_launch(void* const* d_in, const int* in_sizes, int n_in,
                              void* d_out, int out_size, void* d_ws, size_t ws_size,
                              hipStream_t stream) {
    const float* Kmat = (const float*)d_in[0];   // 8192 x 8192
    const float* y    = (const float*)d_in[1];   // 8192 x 1
    const float* Z    = (const float*)d_in[2];   // 8192 x 100

    const size_t colsz = (size_t)MPAD * NN;
    float* ws   = (float*)d_ws;
    float* X    = ws;                  ws += colsz;
    float* R    = ws;                  ws += colsz;
    float* P    = ws;                  ws += colsz;
    float* V    = ws;                  ws += colsz;
    float* Vp   = ws;                  ws += colsz * SPLITK;
    float* rs   = ws;                  ws += MPAD;
    float* rsn  = ws;                  ws += MPAD;
    float* pv   = ws;                  ws += MPAD;
    float* alph = ws;                  ws += (size_t)PITER * MPAD;
    float* bet  = ws;                  ws += (size_t)PITER * MPAD;
    float* quad = ws;                  ws += TPROBE;

    gp_init<<<MPAD, 256, 0, stream>>>(y, Z, X, R, P, rs);

    for (int it = 0; it < PITER; ++it) {
        gp_gemm_wmma<<<dim3(NN / ROWS, SPLITK), TPB, 0, stream>>>(Kmat, P, Vp);
        gp_pv<<<MPAD, 256, 0, stream>>>(Vp, P, V, pv);
        gp_xr<<<MPAD, 256, 0, stream>>>(X, R, P, V, rs, pv, rsn, alph, it);
        gp_pupd<<<MPAD, 256, 0, stream>>>(P, R, rs, rsn, bet, it);
    }

    gp_slq<<<1, 128, 0, stream>>>(alph, bet, quad);
    gp_final<<<1, 256, 0, stream>>>(y, X, quad, (float*)d_out);
}